// CTCLayer_65816078844521
// MI455X (gfx1250) — compile-verified
//
#include <hip/hip_runtime.h>
#include <hip/hip_bf16.h>

// ---------------------------------------------------------------------------
// CTC forward loss (keras ctc_batch_cost semantics), B=256 T=512 C=128 L=32.
// One wave32 per batch element; lane l owns extended states 2l (blank) and
// 2l+1 (label l); lane 31 additionally owns the trailing blank state 64.
// Per time step the only cross-lane dependency is prev-lane's odd alpha ->
// a single __shfl_up. All DP math in log2 domain (v_exp_f32/v_log_f32 native).
// y_pred rows are DMA'd Global->LDS with CDNA5 async copies (ASYNCcnt),
// 6 groups x 8 rows in flight, 8-slot LDS ring buffer.
// ---------------------------------------------------------------------------

#define NEGI  (-1.0e30f)
#define EPSV  (1.0e-7f)
#define LN2F  (0.69314718055994530942f)

typedef int v4i __attribute__((ext_vector_type(4)));
typedef __attribute__((address_space(3))) v4i* lds_v4i_ptr;

#if defined(__has_builtin)
#  if __has_builtin(__builtin_amdgcn_global_load_async_to_lds_b128)
#    define HAVE_ASYNC_COPY 1
#    pragma message("CDNA5 path: __builtin_amdgcn_global_load_async_to_lds_b128 available")
#  else
#    pragma message("CDNA5 path: async-to-LDS builtin NOT available, sync fallback")
#  endif
#  if __has_builtin(__builtin_amdgcn_s_wait_asynccnt)
#    define HAVE_WAIT_ASYNC 1
#    pragma message("CDNA5 path: __builtin_amdgcn_s_wait_asynccnt available")
#  endif
#endif

constexpr int kT    = 512;   // time steps
constexpr int kC    = 128;   // classes (blank = kC-1)
constexpr int kL    = 32;    // label length (== warp size: one label per lane)
constexpr int kG    = 8;     // rows (time steps) per staging group
constexpr int kNBUF = 8;     // LDS ring slots   (8 * 8 * 512B = 32 KB)
constexpr int kPRE  = 6;     // groups in flight (48 async ops < 63 ASYNCcnt max)

// copy one y_pred row (128 f32 = 512B) into LDS: 32 lanes x 16B
__device__ __forceinline__ void copy_row(const float* __restrict__ grow,
                                         float* lrow, int lane) {
#if defined(HAVE_ASYNC_COPY)
  __builtin_amdgcn_global_load_async_to_lds_b128(
      (v4i*)(grow + lane * 4),
      (lds_v4i_ptr)(lrow + lane * 4),
      /*offset=*/0, /*cpol=*/0);
#else
  ((float4*)lrow)[lane] = ((const float4*)grow)[lane];
#endif
}

__device__ __forceinline__ void wait_groups_le() {
  // wait until at most (kPRE-1) groups = (kPRE-1)*kG async ops outstanding
#if defined(HAVE_WAIT_ASYNC)
  __builtin_amdgcn_s_wait_asynccnt((kPRE - 1) * kG);
#elif defined(HAVE_ASYNC_COPY)
  asm volatile("s_wait_asynccnt %0" ::"n"((kPRE - 1) * kG) : "memory");
#else
  // sync fallback: compiler-inserted DS/LOAD waits already order LDS traffic
#endif
}

// one DP step, log2 domain.  a0 = alpha[2l] (blank), a1 = alpha[2l+1] (label),
// a2 = alpha[64] (valid on lane 31 only; computed harmlessly everywhere).
__device__ __forceinline__ void ctc_step(float& a0, float& a1, float& a2,
                                         float lpB, float lpL, bool skip,
                                         int lane) {
  float p1 = __shfl_up(a1, 1u, 32);          // prev lane's alpha[2l-1]
  p1 = (lane == 0) ? NEGI : p1;

  // even (blank) state s=2l: lse(alpha[s], alpha[s-1]) + lpB
  float m0  = fmaxf(a0, p1);
  float s0  = __builtin_amdgcn_exp2f(a0 - m0) + __builtin_amdgcn_exp2f(p1 - m0);
  float na0 = m0 + __builtin_amdgcn_logf(s0) + lpB;

  // odd (label) state s=2l+1: lse(alpha[s], alpha[s-1], skip?alpha[s-2])
  float c   = skip ? p1 : NEGI;
  float m1  = fmaxf(fmaxf(a1, a0), c);
  float s1  = __builtin_amdgcn_exp2f(a1 - m1) + __builtin_amdgcn_exp2f(a0 - m1) +
              __builtin_amdgcn_exp2f(c - m1);
  float na1 = m1 + __builtin_amdgcn_logf(s1) + lpL;

  // trailing blank s=64 (lane 31): lse(alpha[64], alpha[63]) + lpB
  float m2  = fmaxf(a2, a1);
  float s2  = __builtin_amdgcn_exp2f(a2 - m2) + __builtin_amdgcn_exp2f(a1 - m2);
  float na2 = m2 + __builtin_amdgcn_logf(s2) + lpB;

  a0 = na0; a1 = na1; a2 = na2;
}

__global__ __launch_bounds__(32)
void ctc_alpha_kernel(const int* __restrict__ y_true,
                      const float* __restrict__ y_pred,
                      float* __restrict__ out) {
  __shared__ float smem[kNBUF * kG * kC];    // 32 KB ring of staged rows

  const int lane = threadIdx.x;              // 0..31
  const int b    = blockIdx.x;
  const float* __restrict__ ypb = y_pred + (size_t)b * kT * kC;

  const int  lab  = y_true[b * kL + lane];                       // label l
  const int  labm = (lane > 0) ? y_true[b * kL + lane - 1] : -1; // label l-1
  const bool skip = (lane > 0) && (lab != labm);

  // ---- t = 0 init (only states 0 and 1 live, per keras semantics) ----
  float a0 = NEGI, a1 = NEGI, a2 = NEGI;
  {
    float lpB0 = __builtin_amdgcn_logf(ypb[kC - 1] + EPSV);
    float lpL0 = __builtin_amdgcn_logf(ypb[lab] + EPSV);
    if (lane == 0) { a0 = lpB0; a1 = lpL0; }
  }

  constexpr int kNSTEP = kT - 1;                 // 511 remaining steps
  constexpr int kNG    = (kNSTEP + kG - 1) / kG; // 64 groups

  auto issue = [&](int g) {
    const int base_t = 1 + g * kG;
    float* dst = &smem[(g & (kNBUF - 1)) * kG * kC];
#pragma unroll
    for (int k = 0; k < kG; ++k) {
      int t = base_t + k;
      if (t > kT - 1) t = kT - 1;                // tail: duplicate last row
      copy_row(ypb + (size_t)t * kC, dst + k * kC, lane);
    }
  };

  // prologue: fill the pipeline
  for (int g = 0; g < kPRE; ++g) issue(g);

  for (int g = 0; g < kNG; ++g) {
    wait_groups_le();                            // group g fully landed in LDS
    const float* src = &smem[(g & (kNBUF - 1)) * kG * kC];
#pragma unroll
    for (int k = 0; k < kG; ++k) {
      const int t = 1 + g * kG + k;
      if (t <= kT - 1) {
        float lpB = __builtin_amdgcn_logf(src[k * kC + (kC - 1)] + EPSV);
        float lpL = __builtin_amdgcn_logf(src[k * kC + lab] + EPSV);
        ctc_step(a0, a1, a2, lpB, lpL, skip, lane);
      }
    }
    if (g + kPRE < kNG) issue(g + kPRE);         // refill slot read >=2 iters ago
  }

  // ---- finalize: loss = -(lse(alpha[63], alpha[64])) , convert log2->ln ----
  if (lane == 31) {
    float m   = fmaxf(a1, a2);
    float ll2 = m + __builtin_amdgcn_logf(__builtin_amdgcn_exp2f(a1 - m) +
                                          __builtin_amdgcn_exp2f(a2 - m));
    out[b] = -(ll2 * LN2F);
  }
}

extern "C" void kernel_launch(void* const* d_in, const int* in_sizes, int n_in,
                              void* d_out, int out_size, void* d_ws, size_t ws_size,
                              hipStream_t stream) {
  (void)in_sizes; (void)n_in; (void)d_ws; (void)ws_size;
  const int*   y_true = (const int*)d_in[0];    // [B, L] int32
  const float* y_pred = (const float*)d_in[1];  // [B, T, C] f32 softmax probs
  float*       out    = (float*)d_out;          // [B, 1] f32
  const int B = out_size;                       // 256
  hipLaunchKernelGGL(ctc_alpha_kernel, dim3(B), dim3(32), 0, stream,
                     y_true, y_pred, out);
}